// RMSELoss_20985210208413
// MI455X (gfx1250) — compile-verified
//
#include <hip/hip_runtime.h>
#include <math.h>

// ---------------------------------------------------------------------------
// Weighted-RMSE reduction for MI455X (gfx1250, wave32).
// Memory-bound: 128 MB streamed -> ~5.5 us floor at 23.3 TB/s.
// Cross-lane reductions use V_WMMA_F32_16X16X4_F32 (CDNA5 matrix path).
// ---------------------------------------------------------------------------

typedef __attribute__((ext_vector_type(2))) float v2f;
typedef __attribute__((ext_vector_type(8))) float v8f;

#define NBLOCKS 2048
#define NTHREADS 256
#define NWAVES (NTHREADS / 32)

// Piecewise weight: d^3 if d>=0.5 ; d^4 if d<0 ; 1 otherwise. Branchless.
__device__ __forceinline__ float weight_term(float d) {
    float d2 = d * d;
    float early = d2 * d;    // d^3
    float late  = d2 * d2;   // d^4
    float v = (d < 0.0f) ? late : 1.0f;
    v = (d >= 0.5f) ? early : v;
    return v;
}

// Wave-level sum via V_WMMA_F32_16X16X4_F32.
// A[M][K]: lane L<16 supplies A[L][0..1], lane L>=16 supplies A[L-16][2..3].
// With a = {p, 0} and B = all ones, C = 0:
//   D[i][j] = p_i + p_{i+16}   (replicated across all columns j)
// Lanes j<16 hold D rows 0..7, lanes j>=16 hold rows 8..15, so
// sum(own 8 regs) + shfl_xor(.,16) == full 32-lane sum in every lane.
// Requires EXEC == all ones (call sites are uniform, full blocks).
__device__ __forceinline__ float wave_sum_wmma(float p) {
    v2f a; a.x = p;    a.y = 0.0f;
    v2f b; b.x = 1.0f; b.y = 1.0f;
    v8f c = {};
    v8f d = __builtin_amdgcn_wmma_f32_16x16x4_f32(
        /*neg_a=*/false, a, /*neg_b=*/false, b,
        /*c_mod=*/(short)0, c, /*reuse_a=*/false, /*reuse_b=*/false);
    float s = ((d[0] + d[1]) + (d[2] + d[3])) + ((d[4] + d[5]) + (d[6] + d[7]));
    s += __shfl_xor(s, 16, 32);
    return s;
}

// Block-level sum (deterministic: no atomics). Valid result in thread 0.
__device__ __forceinline__ float block_sum(float p) {
    __shared__ float lds[NWAVES];
    const int lane = threadIdx.x & 31;
    const int wave = threadIdx.x >> 5;
    float ws = wave_sum_wmma(p);
    if (lane == 0) lds[wave] = ws;
    __syncthreads();
    float tot = 0.0f;
    if (threadIdx.x == 0) {
#pragma unroll
        for (int i = 0; i < NWAVES; ++i) tot += lds[i];
    }
    return tot;
}

__global__ void __launch_bounds__(NTHREADS)
rmse_partial_kernel(const float* __restrict__ x,   // inputs
                    const float* __restrict__ t,   // targets
                    float* __restrict__ partials,
                    int n) {
    const int n4 = n >> 2;
    const float4* __restrict__ x4 = (const float4*)x;
    const float4* __restrict__ t4 = (const float4*)t;

    // 4 independent accumulators -> short FP dep chains, deterministic order.
    float a0 = 0.0f, a1 = 0.0f, a2 = 0.0f, a3 = 0.0f;

    const int stride = gridDim.x * blockDim.x;
    for (int i = blockIdx.x * blockDim.x + threadIdx.x; i < n4; i += stride) {
        float4 xv = x4[i];          // global_load_b128
        float4 tv = t4[i];          // global_load_b128
        a0 += weight_term(tv.x - xv.x);
        a1 += weight_term(tv.y - xv.y);
        a2 += weight_term(tv.z - xv.z);
        a3 += weight_term(tv.w - xv.w);
    }
    float acc = (a0 + a1) + (a2 + a3);

    // Scalar tail (n not multiple of 4) handled by one thread; reconverges
    // before the WMMA reduction so EXEC is all-1s there.
    if (blockIdx.x == 0 && threadIdx.x == 0) {
        for (int i = n4 << 2; i < n; ++i) acc += weight_term(t[i] - x[i]);
    }

    float tot = block_sum(acc);
    if (threadIdx.x == 0) partials[blockIdx.x] = tot;
}

__global__ void __launch_bounds__(NTHREADS)
rmse_final_kernel(const float* __restrict__ partials, int nparts,
                  float* __restrict__ out, int n) {
    float acc = 0.0f;
    for (int i = threadIdx.x; i < nparts; i += NTHREADS) acc += partials[i];
    float tot = block_sum(acc);
    if (threadIdx.x == 0) {
        out[0] = (float)sqrt((double)tot / (double)n);
    }
}

extern "C" void kernel_launch(void* const* d_in, const int* in_sizes, int n_in,
                              void* d_out, int out_size, void* d_ws, size_t ws_size,
                              hipStream_t stream) {
    const float* x = (const float*)d_in[0];   // inputs  [N,1] f32
    const float* t = (const float*)d_in[1];   // targets [N,1] f32
    float* out = (float*)d_out;               // scalar f32
    float* partials = (float*)d_ws;
    const int n = in_sizes[0];

    int nblocks = NBLOCKS;
    // Defensive: never exceed the provided workspace (deterministic per-size).
    if (ws_size < (size_t)nblocks * sizeof(float)) {
        nblocks = (int)(ws_size / sizeof(float));
        if (nblocks < 1) nblocks = 1;
    }

    rmse_partial_kernel<<<nblocks, NTHREADS, 0, stream>>>(x, t, partials, n);
    rmse_final_kernel<<<1, NTHREADS, 0, stream>>>(partials, nblocks, out, n);
}